// CorrectAndSmoothHeterophily_8108898254941
// MI455X (gfx1250) — compile-verified
//
#include <hip/hip_runtime.h>
#include <cstdint>

#define C_CLS 40
#define ALPHA_CORR 0.979f
#define ALPHA_SMOOTH 0.756f
#define L_CORR 10
#define L_SMOOTH 10
#define EPB 1024   // edges staged per block

typedef float v2f __attribute__((ext_vector_type(2)));
typedef float v8f __attribute__((ext_vector_type(8)));

// ---------------- CDNA5 async global->LDS staging --------------------------
__device__ __forceinline__ void async_b128_to_lds(uint32_t lds_byte_off, const void* gaddr) {
    // VDST = VGPR holding LDS byte offset, VADDR = 64-bit global address, GV mode
    asm volatile("global_load_async_to_lds_b128 %0, %1, off"
                 :: "v"(lds_byte_off), "v"(gaddr) : "memory");
}
__device__ __forceinline__ void wait_asynccnt0() {
    asm volatile("s_wait_asynccnt 0" ::: "memory");
}
__device__ __forceinline__ uint32_t lds_off_of(const void* p) {
    // low 32 bits of a generic pointer into LDS == LDS byte address
    return (uint32_t)(uintptr_t)p;
}

// ---------------- one-shot kernels -----------------------------------------
__global__ void k_degree(const int* __restrict__ dst, float* __restrict__ deg, int E) {
    int e = blockIdx.x * blockDim.x + threadIdx.x;
    if (e < E) atomicAdd(&deg[dst[e]], 1.0f);
}

__global__ void k_dinv(const float* __restrict__ deg, float* __restrict__ dinv, int N) {
    int i = blockIdx.x * blockDim.x + threadIdx.x;
    if (i < N) {
        float d = deg[i];
        dinv[i] = (d > 0.0f) ? rsqrtf(fmaxf(d, 1.0f)) : 0.0f;
    }
}

__global__ void k_norm(const int* __restrict__ src, const int* __restrict__ dst,
                       const float* __restrict__ dinv, float* __restrict__ norm, int E) {
    int e = blockIdx.x * blockDim.x + threadIdx.x;
    if (e < E) norm[e] = dinv[src[e]] * dinv[dst[e]];
}

// error = mask ? onehot - y_soft : 0 ; also counts numel into scal[1]
// 320 threads = 8 rows x 40 classes (10 waves/block)
__global__ __launch_bounds__(320)
void k_error(const int* __restrict__ yt, const float* __restrict__ ys,
             const int* __restrict__ mask, float* __restrict__ err0,
             float* __restrict__ x, float* scal, int N) {
    int r = blockIdx.x * 8 + threadIdx.x / C_CLS;
    int c = threadIdx.x % C_CLS;
    if (r >= N) return;
    int m = mask[r];
    float oh = (yt[r] == c) ? 1.0f : 0.0f;
    float v  = m ? (oh - ys[(size_t)r * C_CLS + c]) : 0.0f;
    err0[(size_t)r * C_CLS + c] = v;
    x[(size_t)r * C_CLS + c]    = v;
    if (c == 0 && m) atomicAdd(&scal[1], 1.0f);
}

// ---------------- WMMA row-|sum| reduction ---------------------------------
// One wave reduces a 16-row x 40-col tile: D = |A|(16x4) * ones(4x16), K-chunked
// 10x.  Every column of D holds the row sum; lane 0 extracts rows 0..7,
// lane 16 extracts rows 8..15 (per the 16x16 f32 C/D VGPR layout).
// Fast path (tile fully in range, wave-uniform test): 10 unconditional b64
// loads issued ahead of the WMMA chain -> one loadcnt wait, no EXEC churn.
__global__ void k_rowsum_abs_wmma(const float* __restrict__ x, float* __restrict__ rowout,
                                  float* __restrict__ total, int N) {
    int wave = (blockIdx.x * blockDim.x + threadIdx.x) >> 5;
    int lane = threadIdx.x & 31;
    int row0 = wave << 4;
    if (row0 >= N) return;                    // wave-uniform; EXEC stays all-1s
    int m  = lane & 15;
    int kh = (lane >> 4) << 1;                // lanes 0-15: K=0,1 ; lanes 16-31: K=2,3
    int r  = row0 + m;
    v2f b; b.x = 1.0f; b.y = 1.0f;            // all-ones B => D[m,n] = sum_k A[m,k]
    v8f acc = {};

    if (row0 + 16 <= N) {                     // uniform full-tile fast path
        const float2* rp = (const float2*)(x + (size_t)r * C_CLS + kh);  // 8B aligned
        float2 va[C_CLS / 4];
#pragma unroll
        for (int j = 0; j < C_CLS / 4; ++j) va[j] = rp[j * 2];           // global_load_b64
#pragma unroll
        for (int j = 0; j < C_CLS / 4; ++j) {
            v2f a;
            a.x = __builtin_fabsf(va[j].x);
            a.y = __builtin_fabsf(va[j].y);
            acc = __builtin_amdgcn_wmma_f32_16x16x4_f32(false, a, false, b,
                                                        (short)0, acc, false, false);
        }
    } else {                                  // ragged tail tile (at most one wave)
        bool rv = (r < N);
        const float* rp = x + (size_t)r * C_CLS;
#pragma unroll
        for (int k0 = 0; k0 < C_CLS; k0 += 4) {
            v2f a;
            a.x = rv ? __builtin_fabsf(rp[k0 + kh + 0]) : 0.0f;
            a.y = rv ? __builtin_fabsf(rp[k0 + kh + 1]) : 0.0f;
            acc = __builtin_amdgcn_wmma_f32_16x16x4_f32(false, a, false, b,
                                                        (short)0, acc, false, false);
        }
    }

    if ((lane & 15) == 0) {
        int rbase = row0 + ((lane >> 4) << 3);
        float s = 0.0f;
#pragma unroll
        for (int j = 0; j < 8; ++j) {
            float v = acc[j];
            if (rowout && (rbase + j) < N) rowout[rbase + j] = v;
            s += v;
        }
        if (total) atomicAdd(total, s);
    }
}

// ---------------- edge scatter (the hot loop) ------------------------------
__global__ __launch_bounds__(256)
void k_scatter(const int* __restrict__ src, const int* __restrict__ dst,
               const float* __restrict__ w, const float* __restrict__ x,
               float* __restrict__ agg, int E) {
    __shared__ int   sS[EPB];
    __shared__ int   sD[EPB];
    __shared__ float sW[EPB];
    int base = blockIdx.x * EPB;
    int n = E - base; if (n > EPB) n = EPB;
    int tid = threadIdx.x;

    int i = tid * 4;                          // 256 lanes x 16B = 4KB per array
    if (i < n) {
        async_b128_to_lds(lds_off_of(&sS[i]), src + base + i);
        async_b128_to_lds(lds_off_of(&sD[i]), dst + base + i);
        async_b128_to_lds(lds_off_of(&sW[i]), w   + base + i);
    }
    wait_asynccnt0();
    __syncthreads();

    for (int e = tid; e < n; e += 256) {
        int   s  = sS[e];
        int   d  = sD[e];
        float ww = sW[e];
        const float4* xv = (const float4*)(x + (size_t)s * C_CLS);  // 160B row, 16B aligned
        float* ar = agg + (size_t)d * C_CLS;
#pragma unroll
        for (int j = 0; j < C_CLS / 4; ++j) {
            float4 v = xv[j];                                   // global_load_b128 (L2 hit)
            atomicAdd(&ar[4 * j + 0], v.x * ww);                // global_atomic_add_f32
            atomicAdd(&ar[4 * j + 1], v.y * ww);
            atomicAdd(&ar[4 * j + 2], v.z * ww);
            atomicAdd(&ar[4 * j + 3], v.w * ww);
        }
    }
}

// ---------------- post steps ----------------------------------------------
__global__ void k_post_clip(const float* __restrict__ agg, const float* __restrict__ y0,
                            float* __restrict__ x, int NC) {
    int i = blockIdx.x * blockDim.x + threadIdx.x;
    if (i < NC) {
        float v = ALPHA_CORR * agg[i] + (1.0f - ALPHA_CORR) * y0[i];
        x[i] = fminf(1.0f, fmaxf(-1.0f, v));
    }
}

__global__ __launch_bounds__(320)
void k_correct(const int* __restrict__ yt, const float* __restrict__ ys,
               const int* __restrict__ mask, const float* sm,
               const float* __restrict__ denom, const float* __restrict__ scal,
               float* __restrict__ y0, float* x, int N) {
    int r = blockIdx.x * 8 + threadIdx.x / C_CLS;
    int c = threadIdx.x % C_CLS;
    if (r >= N) return;
    float sigma = scal[0] / scal[1];
    float scale = sigma / denom[r];
    if (__builtin_isinf(scale) || scale > 1000.0f) scale = 1.0f;   // NaN passes through
    float oh = (yt[r] == c) ? 1.0f : 0.0f;
    float v  = mask[r] ? oh : (ys[(size_t)r * C_CLS + c] + scale * sm[(size_t)r * C_CLS + c]);
    y0[(size_t)r * C_CLS + c] = v;
    x[(size_t)r * C_CLS + c]  = v;
}

__global__ void k_post_softmax(const float* __restrict__ agg, const float* __restrict__ y0,
                               float* __restrict__ x, int N) {
    int r = blockIdx.x * blockDim.x + threadIdx.x;
    if (r >= N) return;
    const float* a = agg + (size_t)r * C_CLS;
    const float* y = y0  + (size_t)r * C_CLS;
    float t[C_CLS];
    float mx = -3.0e38f;
#pragma unroll
    for (int c = 0; c < C_CLS; ++c) {
        float v = ALPHA_SMOOTH * a[c] + (1.0f - ALPHA_SMOOTH) * y[c];
        t[c] = v;
        mx = fmaxf(mx, v);
    }
    float s = 0.0f;
#pragma unroll
    for (int c = 0; c < C_CLS; ++c) {
        float e = __expf(t[c] - mx);
        t[c] = e;
        s += e;
    }
    float inv = 1.0f / s;
    float* o = x + (size_t)r * C_CLS;
#pragma unroll
    for (int c = 0; c < C_CLS; ++c) o[c] = t[c] * inv;
}

// ---------------- driver ---------------------------------------------------
extern "C" void kernel_launch(void* const* d_in, const int* in_sizes, int n_in,
                              void* d_out, int out_size, void* d_ws, size_t ws_size,
                              hipStream_t stream) {
    (void)n_in; (void)out_size;
    const int*   y_true = (const int*)d_in[0];
    const float* y_soft = (const float*)d_in[1];
    const int*   mask   = (const int*)d_in[2];
    const int*   ei     = (const int*)d_in[3];

    int N = in_sizes[0];
    int E = in_sizes[3] / 2;
    const int* src = ei;
    const int* dst = ei + E;
    size_t NC = (size_t)N * C_CLS;

    char* ws = (char*)d_ws;
    size_t off = 0;
    auto take = [&](size_t bytes) -> char* {
        char* p = ws + off;
        off = (off + bytes + 255) & ~(size_t)255;
        return p;
    };
    float* scal  = (float*)take(256);          // [0]=sigma_sum [1]=numel
    float* deg   = (float*)take((size_t)N * 4);
    float* dinv  = (float*)take((size_t)N * 4);
    float* denom = (float*)take((size_t)N * 4);
    float* norm  = (float*)take((size_t)E * 4);
    float* err0  = (float*)take(NC * 4);
    float* x     = (float*)take(NC * 4);
    float* y0    = (float*)take(NC * 4);
    float* agg   = (float*)take(NC * 4);
    if (off > ws_size) return;                 // workspace too small: bail deterministically

    const int tb = 256;
    int eb  = (E + tb - 1) / tb;
    int nb  = (N + tb - 1) / tb;
    int ncb = (int)((NC + tb - 1) / tb);
    int rowb = (N + 7) / 8;                    // 8 rows per 320-thread block
    int tiles = (N + 15) / 16;
    int wmma_blocks = (tiles + 7) / 8;         // 8 waves/block, 1 tile/wave
    int scat_blocks = (E + EPB - 1) / EPB;

    hipMemsetAsync(scal, 0, 256, stream);
    hipMemsetAsync(deg,  0, (size_t)N * 4, stream);

    k_degree<<<eb, tb, 0, stream>>>(dst, deg, E);
    k_dinv  <<<nb, tb, 0, stream>>>(deg, dinv, N);
    k_norm  <<<eb, tb, 0, stream>>>(src, dst, dinv, norm, E);
    k_error <<<rowb, 320, 0, stream>>>(y_true, y_soft, mask, err0, x, scal, N);
    k_rowsum_abs_wmma<<<wmma_blocks, 256, 0, stream>>>(err0, nullptr, &scal[0], N);

    for (int it = 0; it < L_CORR; ++it) {
        hipMemsetAsync(agg, 0, NC * 4, stream);
        k_scatter  <<<scat_blocks, 256, 0, stream>>>(src, dst, norm, x, agg, E);
        k_post_clip<<<ncb, tb, 0, stream>>>(agg, err0, x, (int)NC);
    }

    k_rowsum_abs_wmma<<<wmma_blocks, 256, 0, stream>>>(x, denom, nullptr, N);
    k_correct<<<rowb, 320, 0, stream>>>(y_true, y_soft, mask, x, denom, scal, y0, x, N);

    for (int it = 0; it < L_SMOOTH; ++it) {
        hipMemsetAsync(agg, 0, NC * 4, stream);
        k_scatter     <<<scat_blocks, 256, 0, stream>>>(src, dst, norm, x, agg, E);
        k_post_softmax<<<(N + 127) / 128, 128, 0, stream>>>(agg, y0, x, N);
    }

    hipMemcpyAsync(d_out, x, NC * 4, hipMemcpyDeviceToDevice, stream);
}